// ItemAttention_63376537420339
// MI455X (gfx1250) — compile-verified
//
#include <hip/hip_runtime.h>
#include <cstdint>
#include <cstddef>

typedef __attribute__((ext_vector_type(2))) float v2f;
typedef __attribute__((ext_vector_type(8))) float v8f;

#define B_ 4096
#define M_ 200
#define E_ 128
#define R_ 64
#define ROWSTRIDE 132          // padded LDS row stride (floats): bank = (e + 4m) % 64, conflict-free
#define MT 13                  // ceil(200/16) m-tiles
#define SM_INTER_ROWS 208      // 13 tiles * 16 rows (rows 200..207 zero-filled)

// shared memory layout, in floats
#define OFF_INTER 0
#define OFF_C     (SM_INTER_ROWS * ROWSTRIDE)   // 27456
#define OFF_UW    (OFF_C + E_)
#define OFF_SC    (OFF_UW + E_)
#define OFF_RED   (OFF_SC + 256)
#define SMEM_FLOATS (OFF_RED + 256)             // 28224
#define SMEM_BYTES  (SMEM_FLOATS * 4)           // 112896 bytes

// ---- tiny reduction kernel: w_sum[e] = sum_r key[r,e]*val[r,e] ----
__global__ void wsum_kernel(const float* __restrict__ key,
                            const float* __restrict__ val,
                            float* __restrict__ wsum) {
  int e = threadIdx.x;            // 128 threads
  float s = 0.f;
#pragma unroll 8
  for (int r = 0; r < R_; ++r)
    s = fmaf(key[r * E_ + e], val[r * E_ + e], s);
  wsum[e] = s;
}

// ---- main kernel: one workgroup (8 wave32) per batch element ----
__global__ void __launch_bounds__(256) attn_kernel(
    const float* __restrict__ user, const float* __restrict__ item,
    const float* __restrict__ inter, const float* __restrict__ wsum,
    float* __restrict__ out) {
  extern __shared__ float smem[];
  float* sm_inter = smem + OFF_INTER;   // 208 x 132 padded tile of inter[b]
  float* sm_c     = smem + OFF_C;       // user*item*w_sum
  float* sm_uw    = smem + OFF_UW;      // user*w_sum
  float* sm_sc    = smem + OFF_SC;      // scores -> exp(scores)
  float* sm_red   = smem + OFF_RED;     // reduction scratch

  const int t    = threadIdx.x;
  const int b    = blockIdx.x;
  const int lane = t & 31;
  const int wid  = t >> 5;

  // --- async-stream inter[b] (200x128 f32 = 100KB) into padded LDS tile ---
  // Streamed exactly once across the whole grid -> non-temporal hint keeps the
  // 419MB of inter from evicting the small broadcast vectors out of L2.
  const float* gbase = inter + (size_t)b * (M_ * E_);
  for (int i = t; i < M_ * (E_ / 4); i += 256) {   // 6400 16-byte chunks, 25/thread
    int row = i >> 5;
    int c4  = i & 31;
    unsigned lds_addr = (unsigned)(uintptr_t)(sm_inter + row * ROWSTRIDE + c4 * 4);
    unsigned long long gaddr =
        (unsigned long long)(uintptr_t)(gbase + row * E_ + c4 * 4);
    asm volatile("global_load_async_to_lds_b128 %0, %1, off th:TH_LOAD_NT"
                 :: "v"(lds_addr), "v"(gaddr) : "memory");
  }

  // zero-fill the 8 padding rows (m = 200..207) so the last WMMA tile is benign
  for (int i = t; i < (SM_INTER_ROWS - M_) * ROWSTRIDE; i += 256)
    sm_inter[M_ * ROWSTRIDE + i] = 0.f;

  // per-feature vectors
  if (t < E_) {
    float w  = wsum[t];
    float u  = user[(size_t)b * E_ + t];
    float it = item[(size_t)b * E_ + t];
    sm_uw[t] = u * w;
    sm_c[t]  = u * w * it;
  }

  asm volatile("s_wait_asynccnt 0x0" ::: "memory");
  __syncthreads();

  // --- stage 1: scores[m] = sum_e inter[m,e] * c[e] via v_wmma_f32_16x16x4_f32 ---
  // A[m,k] = inter[m, e0+k]; B[k,n] = c[e0+k] broadcast over n -> every D column
  // holds the partial score. 32 K-steps, software-pipelined A loads (8-deep
  // register ring) + two independent accumulator chains to break the D->C
  // latency chain.
  const int koff = (lane < 16) ? 0 : 2;   // K held by this half-wave in VGPR0/1
  for (int tile = wid; tile < MT; tile += 8) {
    int m = tile * 16 + (lane & 15);
    const float* arow = sm_inter + m * ROWSTRIDE + koff;
    const float* brow = sm_c + koff;

    v2f abuf[8];
#pragma unroll
    for (int k = 0; k < 8; ++k) {         // prologue: first 8 K-steps in flight
      abuf[k].x = arow[4 * k];
      abuf[k].y = arow[4 * k + 1];
    }

    v8f acc0 = {0.f, 0.f, 0.f, 0.f, 0.f, 0.f, 0.f, 0.f};
    v8f acc1 = {0.f, 0.f, 0.f, 0.f, 0.f, 0.f, 0.f, 0.f};
#pragma unroll
    for (int k = 0; k < 32; ++k) {
      v2f a = abuf[k & 7];
      if (k + 8 < 32) {                   // refill the ring 8 steps ahead
        abuf[k & 7].x = arow[4 * (k + 8)];
        abuf[k & 7].y = arow[4 * (k + 8) + 1];
      }
      v2f bb;
      bb.x = brow[4 * k];
      bb.y = brow[4 * k + 1];
      if (k & 1)
        acc1 = __builtin_amdgcn_wmma_f32_16x16x4_f32(
            false, a, false, bb, (short)0, acc1, false, false);
      else
        acc0 = __builtin_amdgcn_wmma_f32_16x16x4_f32(
            false, a, false, bb, (short)0, acc0, false, false);
    }

    // D layout: VGPR v holds M=v (lanes 0-15) / M=v+8 (lanes 16-31); take column N=0
    if ((lane & 15) == 0) {
      int base = tile * 16 + ((lane >> 4) << 3);
#pragma unroll
      for (int v = 0; v < 8; ++v) sm_sc[base + v] = acc0[v] + acc1[v];
    }
  }
  __syncthreads();

  // --- softmax over m = 0..199 ---
  float sval = (t < M_) ? sm_sc[t] : -__builtin_inff();
  sm_red[t] = sval;
  __syncthreads();
  for (int s = 128; s > 0; s >>= 1) {
    if (t < s) sm_red[t] = fmaxf(sm_red[t], sm_red[t + s]);
    __syncthreads();
  }
  float mx = sm_red[0];
  __syncthreads();
  float ex = (t < M_) ? __expf(sval - mx) : 0.f;
  sm_sc[t]  = ex;           // unnormalized attn (0 for m >= 200)
  sm_red[t] = ex;
  __syncthreads();
  for (int s = 128; s > 0; s >>= 1) {
    if (t < s) sm_red[t] += sm_red[t + s];
    __syncthreads();
  }
  float inv = 1.0f / sm_red[0];
  __syncthreads();

  // --- stage 2: out[e] = uw[e] * inv * sum_m attn[m] * inter[m,e] ---
  int e    = t & (E_ - 1);
  int half = t >> 7;                 // two thread-groups split the m range
  const float* col = sm_inter + e;
  float part = 0.f;
  int m0 = half * (M_ / 2);
#pragma unroll 4
  for (int m = m0; m < m0 + M_ / 2; ++m)
    part = fmaf(sm_sc[m], col[m * ROWSTRIDE], part);
  sm_red[t] = part;
  __syncthreads();
  if (t < E_) {
    float o = sm_uw[t] * (sm_red[t] + sm_red[t + 128]) * inv;
    out[(size_t)b * E_ + t] = o;
  }
}

extern "C" void kernel_launch(void* const* d_in, const int* in_sizes, int n_in,
                              void* d_out, int out_size, void* d_ws, size_t ws_size,
                              hipStream_t stream) {
  const float* user  = (const float*)d_in[0];
  const float* item  = (const float*)d_in[1];
  const float* inter = (const float*)d_in[2];
  const float* key_w = (const float*)d_in[3];
  const float* val_w = (const float*)d_in[4];
  float* wsum = (float*)d_ws;       // 128 floats of scratch
  float* out  = (float*)d_out;

  (void)in_sizes; (void)n_in; (void)out_size; (void)ws_size;

  static_cast<void>(hipFuncSetAttribute(
      (const void*)attn_kernel,
      hipFuncAttributeMaxDynamicSharedMemorySize, SMEM_BYTES));

  wsum_kernel<<<1, 128, 0, stream>>>(key_w, val_w, wsum);
  attn_kernel<<<B_, 256, SMEM_BYTES, stream>>>(user, item, inter, wsum, out);
}